// SAGE_7937099563499
// MI455X (gfx1250) — compile-verified
//
#include <hip/hip_runtime.h>

#define N_NODES 50000
#define N_EDGES 800000
#define D 128
#define SROW 132   // LDS row stride (floats): 132 % 64 = 4 -> 16 rows hit 16 distinct banks

typedef float v2f __attribute__((ext_vector_type(2)));
typedef float v8f __attribute__((ext_vector_type(8)));

// ---------------- utility kernels ----------------

__global__ void sage_zero_kernel(float* __restrict__ p, int n) {
    int i = blockIdx.x * blockDim.x + threadIdx.x;
    if (i < n) p[i] = 0.0f;
}

__global__ void sage_count_kernel(const long long* __restrict__ dst,
                                  float* __restrict__ cnt) {
    int e = blockIdx.x * blockDim.x + threadIdx.x;
    if (e < N_EDGES) {
        int d = (int)dst[e];
        __hip_atomic_fetch_add(&cnt[d], 1.0f, __ATOMIC_RELAXED, __HIP_MEMORY_SCOPE_AGENT);
    }
}

__global__ void sage_invcnt_kernel(float* __restrict__ cnt) {
    int i = blockIdx.x * blockDim.x + threadIdx.x;
    if (i < N_NODES) cnt[i] = 1.0f / fmaxf(cnt[i], 1.0f);
}

// ---------------- edge scatter (mean-aggregate numerator) ----------------
// One wave per edge: 32 lanes cover the 128-float feature row.
// x[src] read is one coalesced 512B line; src/dst loads are wave-uniform.
__global__ __launch_bounds__(256) void sage_scatter_kernel(
    const float* __restrict__ x, const long long* __restrict__ src,
    const long long* __restrict__ dst, float* __restrict__ agg) {
    int gid = blockIdx.x * 256 + threadIdx.x;
    int e = gid >> 5;            // edge index (uniform across the wave)
    int c = (gid & 31) * 4;      // 4-float chunk within the feature row
    if (e >= N_EDGES) return;
    long long s = src[e];
    long long d = dst[e];
    const float4 v = *(const float4*)(x + (long)s * D + c);
    float* a = agg + (long)d * D + c;
    __hip_atomic_fetch_add(a + 0, v.x, __ATOMIC_RELAXED, __HIP_MEMORY_SCOPE_AGENT);
    __hip_atomic_fetch_add(a + 1, v.y, __ATOMIC_RELAXED, __HIP_MEMORY_SCOPE_AGENT);
    __hip_atomic_fetch_add(a + 2, v.z, __ATOMIC_RELAXED, __HIP_MEMORY_SCOPE_AGENT);
    __hip_atomic_fetch_add(a + 3, v.w, __ATOMIC_RELAXED, __HIP_MEMORY_SCOPE_AGENT);
}

// ---------------- fused SAGE layer GEMM ----------------
// out[m][:] = relu( (agg[m]*inv[m]) @ Wl + bl + xin[m] @ Wr )
// Block = 256 threads = 8 waves. Block computes a 16-node x 128-col tile.
// Wave w computes cols [16w, 16w+16) with fp32 WMMA 16x16x4, K=128.
__global__ __launch_bounds__(256) void sage_gemm_kernel(
    const float* __restrict__ agg, const float* __restrict__ inv,
    const float* __restrict__ xin,
    const float* __restrict__ Wl, const float* __restrict__ bl,
    const float* __restrict__ Wr, float* __restrict__ out) {
    __shared__ float sAgg[16 * SROW];
    __shared__ float sX[16 * SROW];

    const int tid = threadIdx.x;
    const long nodeBase = (long)blockIdx.x * 16;   // 50000 = 3125*16, no tail

    // ---- stage A-tiles into LDS (agg pre-scaled by 1/deg) ----
    {
        int r = tid >> 4;            // 0..15 node row
        int c = (tid & 15) * 8;      // 0..120 col
        long g = (nodeBase + r) * D + c;
        float iv = inv[nodeBase + r];
        float4 a0 = *(const float4*)(agg + g);
        float4 a1 = *(const float4*)(agg + g + 4);
        float4 x0 = *(const float4*)(xin + g);
        float4 x1 = *(const float4*)(xin + g + 4);
        float* sa = &sAgg[r * SROW + c];
        sa[0] = a0.x * iv; sa[1] = a0.y * iv; sa[2] = a0.z * iv; sa[3] = a0.w * iv;
        sa[4] = a1.x * iv; sa[5] = a1.y * iv; sa[6] = a1.z * iv; sa[7] = a1.w * iv;
        float* sx = &sX[r * SROW + c];
        sx[0] = x0.x; sx[1] = x0.y; sx[2] = x0.z; sx[3] = x0.w;
        sx[4] = x1.x; sx[5] = x1.y; sx[6] = x1.z; sx[7] = x1.w;
    }
    __syncthreads();

    const int lane = tid & 31;
    const int wave = tid >> 5;                // 0..7
    const int m    = lane & 15;               // A row within tile
    const int half = lane >> 4;               // K-half selector
    const int col  = wave * 16 + (lane & 15); // B/out column 0..127

    v8f cA = {};   // agg @ Wl accumulator
    v8f cX = {};   // x   @ Wr accumulator (independent chain for ILP)

#pragma unroll
    for (int k = 0; k < D; k += 4) {
        const int kk = k + half * 2;
        v2f aA = *(const v2f*)&sAgg[m * SROW + kk];
        v2f aX = *(const v2f*)&sX[m * SROW + kk];
        v2f bL = { Wl[kk * D + col], Wl[(kk + 1) * D + col] };
        v2f bR = { Wr[kk * D + col], Wr[(kk + 1) * D + col] };
        cA = __builtin_amdgcn_wmma_f32_16x16x4_f32(false, aA, false, bL,
                                                   (short)0, cA, false, false);
        cX = __builtin_amdgcn_wmma_f32_16x16x4_f32(false, aX, false, bR,
                                                   (short)0, cX, false, false);
    }

    const float bias = bl[col];
#pragma unroll
    for (int r = 0; r < 8; ++r) {
        float v = cA[r] + cX[r] + bias;
        v = fmaxf(v, 0.0f);
        out[(nodeBase + r + half * 8) * D + col] = v;
    }
}

// ---------------- launch ----------------

extern "C" void kernel_launch(void* const* d_in, const int* in_sizes, int n_in,
                              void* d_out, int out_size, void* d_ws, size_t ws_size,
                              hipStream_t stream) {
    const float*     x    = (const float*)d_in[0];
    const long long* eidx = (const long long*)d_in[1];   // int64 in reference
    const float*     Wl0  = (const float*)d_in[2];
    const float*     bl0  = (const float*)d_in[3];
    const float*     Wr0  = (const float*)d_in[4];
    const float*     Wl1  = (const float*)d_in[5];
    const float*     bl1  = (const float*)d_in[6];
    const float*     Wr1  = (const float*)d_in[7];
    const long long* src  = eidx;
    const long long* dst  = eidx + N_EDGES;

    float* agg = (float*)d_ws;                       // 50000*128 f32
    float* h   = agg + (size_t)N_NODES * D;          // 50000*128 f32
    float* cnt = h + (size_t)N_NODES * D;            // 50000 f32 (becomes 1/deg)

    const int ND = N_NODES * D;
    dim3 b256(256);

    // degree counts (shared by both layers)
    sage_zero_kernel<<<(N_NODES + 255) / 256, b256, 0, stream>>>(cnt, N_NODES);
    sage_count_kernel<<<(N_EDGES + 255) / 256, b256, 0, stream>>>(dst, cnt);
    sage_invcnt_kernel<<<(N_NODES + 255) / 256, b256, 0, stream>>>(cnt);

    // ---- layer 0 ----
    sage_zero_kernel<<<(ND + 255) / 256, b256, 0, stream>>>(agg, ND);
    sage_scatter_kernel<<<(N_EDGES * 32) / 256, b256, 0, stream>>>(x, src, dst, agg);
    sage_gemm_kernel<<<N_NODES / 16, b256, 0, stream>>>(agg, cnt, x, Wl0, bl0, Wr0, h);

    // ---- layer 1 ----
    sage_zero_kernel<<<(ND + 255) / 256, b256, 0, stream>>>(agg, ND);
    sage_scatter_kernel<<<(N_EDGES * 32) / 256, b256, 0, stream>>>(h, src, dst, agg);
    sage_gemm_kernel<<<N_NODES / 16, b256, 0, stream>>>(agg, cnt, h, Wl1, bl1, Wr1,
                                                        (float*)d_out);
}